// GATModel_85907935854811
// MI455X (gfx1250) — compile-verified
//
#include <hip/hip_runtime.h>
#include <hip/hip_bf16.h>
#include <math.h>

#define IN_F 128
#define HEADS 8
#define HC 16          // HEADS * OUT_C
#define NEG_SLOPE 0.2f

typedef __attribute__((ext_vector_type(2))) float v2f;
typedef __attribute__((ext_vector_type(8))) float v8f;

// ---- monotone uint encoding of float for atomicMax-based segment max ----
__device__ __forceinline__ unsigned fkey(float f) {
    unsigned u = __float_as_uint(f);
    return u ^ ((unsigned)((int)u >> 31) | 0x80000000u);
}
__device__ __forceinline__ float funkey(unsigned k) {
    return (k & 0x80000000u) ? __uint_as_float(k ^ 0x80000000u)
                             : __uint_as_float(~k);
}

// ============================================================
// Kernel 1: h = x @ W  via V_WMMA_F32_16X16X4_F32
//   one wave -> one 16(row) x 16(col) tile, K=128 in 32 steps of 4.
//   W^T staged in LDS (B frag = one ds_load_b64); D tile staged in
//   LDS and written back as coalesced b128 stores.
// ============================================================
__global__ void __launch_bounds__(256)
gat_gemm_wmma(const float* __restrict__ x,
              const float* __restrict__ W,
              float* __restrict__ h, int N) {
    __shared__ float Wt[HC * IN_F];        // W transposed: Wt[n*128 + k]
    __shared__ float tile[8][16 * 16];     // per-wave D staging

    // stage W^T once per block (coalesced global read, transposed LDS write)
    for (int i = (int)threadIdx.x; i < HC * IN_F; i += (int)blockDim.x) {
        const int k = i >> 4, n = i & 15;  // W is [K=128][N=16] row-major
        Wt[n * IN_F + k] = W[i];
    }
    __syncthreads();

    const int waveInBlk = (int)(threadIdx.x >> 5);
    const int wave = (int)blockIdx.x * ((int)blockDim.x >> 5) + waveInBlk;
    const int lane = (int)(threadIdx.x & 31);
    const int row0 = wave * 16;
    if (row0 >= N) return;                 // wave-uniform: EXEC stays all-ones

    const int m     = lane & 15;           // A: row-in-tile ; B/D: column
    const int khalf = lane >> 4;           // selects K pair {0,1} vs {2,3}

    int arow = row0 + m; if (arow >= N) arow = N - 1;
    const float* __restrict__ xrow = x + (size_t)arow * IN_F;
    const float* wrow = Wt + (size_t)m * IN_F;

    v8f c = {};
    #pragma unroll 8
    for (int k0 = 0; k0 < IN_F; k0 += 4) {
        const int ka = k0 + khalf * 2;
        v2f a = *(const v2f*)(xrow + ka);   // global_load_b64 (A layout per ISA)
        v2f b = *(const v2f*)(wrow + ka);   // ds_load_b64     (B layout per ISA)
        c = __builtin_amdgcn_wmma_f32_16x16x4_f32(
                /*neg_a=*/false, a, /*neg_b=*/false, b,
                /*c_mod=*/(short)0, c, /*reuse_a=*/false, /*reuse_b=*/false);
    }

    // D layout: VGPR i -> row = i + khalf*8, col = m.  Stage tile in LDS.
    float* tl = tile[waveInBlk];
    #pragma unroll
    for (int i = 0; i < 8; ++i)
        tl[(khalf * 8 + i) * 16 + m] = c[i];
    // same-wave DS RAW: compiler inserts s_wait_dscnt

    // coalesced write-back: lane -> row = lane/2, 8 floats (two b128)
    const int r    = lane >> 1;
    const int part = (lane & 1) * 8;
    float4 p0 = *(const float4*)(tl + r * 16 + part);
    float4 p1 = *(const float4*)(tl + r * 16 + part + 4);
    if (row0 + 16 <= N) {                  // wave-uniform fast path
        float* dst = h + (size_t)(row0 + r) * HC + part;
        *(float4*)(dst)     = p0;
        *(float4*)(dst + 4) = p1;
    } else if (row0 + r < N) {
        float* dst = h + (size_t)(row0 + r) * HC + part;
        *(float4*)(dst)     = p0;
        *(float4*)(dst + 4) = p1;
    }
}

// ============================================================
// Kernel 2: per-node attention scalars a_src / a_dst  [N,8]
// ============================================================
__global__ void gat_node_att(const float* __restrict__ h,
                             const float* __restrict__ att_src,
                             const float* __restrict__ att_dst,
                             float* __restrict__ asrc,
                             float* __restrict__ adst, int N) {
    const int n = (int)(blockIdx.x * blockDim.x + threadIdx.x);
    if (n >= N) return;
    const float4* hp = (const float4*)(h + (size_t)n * HC);
    float hv[HC];
    #pragma unroll
    for (int q = 0; q < 4; ++q) {
        float4 v = hp[q];
        hv[4*q+0] = v.x; hv[4*q+1] = v.y; hv[4*q+2] = v.z; hv[4*q+3] = v.w;
    }
    float4 s0, s1, d0, d1;
    #pragma unroll
    for (int hd = 0; hd < 4; ++hd) {
        ((float*)&s0)[hd] = hv[2*hd]   * att_src[2*hd]   + hv[2*hd+1]   * att_src[2*hd+1];
        ((float*)&s1)[hd] = hv[2*hd+8] * att_src[2*hd+8] + hv[2*hd+9]   * att_src[2*hd+9];
        ((float*)&d0)[hd] = hv[2*hd]   * att_dst[2*hd]   + hv[2*hd+1]   * att_dst[2*hd+1];
        ((float*)&d1)[hd] = hv[2*hd+8] * att_dst[2*hd+8] + hv[2*hd+9]   * att_dst[2*hd+9];
    }
    float4* ps = (float4*)(asrc + (size_t)n * HEADS);
    float4* pd = (float4*)(adst + (size_t)n * HEADS);
    ps[0] = s0; ps[1] = s1;
    pd[0] = d0; pd[1] = d1;
}

// ============================================================
// Kernel 3: zero the accumulator region (kmax keys, segsum, agg)
//   vectorized b128 fill (region is 16B-aligned, N*32 dwords)
// ============================================================
__global__ void gat_zero(uint4* __restrict__ p, int n4) {
    const int i = (int)(blockIdx.x * blockDim.x + threadIdx.x);
    if (i < n4) p[i] = make_uint4(0u, 0u, 0u, 0u);
}

// ============================================================
// Kernel 4: segment max over edges (incl. implicit self-loops)
// ============================================================
__global__ void gat_edge_max(const long long* __restrict__ ei, int E, int N,
                             const float* __restrict__ asrc,
                             const float* __restrict__ adst,
                             unsigned* __restrict__ kmax) {
    const int e = (int)(blockIdx.x * blockDim.x + threadIdx.x);
    const int Et = E + N;
    if (e >= Et) return;
    __builtin_prefetch(ei + e + 4096, 0, 0);     // global_prefetch_b8 on index stream
    int s, d;
    if (e < E) { s = (int)ei[e]; d = (int)ei[(size_t)E + e]; }
    else       { s = e - E; d = s; }
    const float4* As = (const float4*)(asrc + (size_t)s * HEADS);
    const float4* Ad = (const float4*)(adst + (size_t)d * HEADS);
    float4 s0 = As[0], s1 = As[1], d0 = Ad[0], d1 = Ad[1];
    float l[HEADS] = { s0.x + d0.x, s0.y + d0.y, s0.z + d0.z, s0.w + d0.w,
                       s1.x + d1.x, s1.y + d1.y, s1.z + d1.z, s1.w + d1.w };
    unsigned* km = kmax + (size_t)d * HEADS;
    #pragma unroll
    for (int hd = 0; hd < HEADS; ++hd) {
        float v = l[hd];
        v = (v > 0.0f) ? v : NEG_SLOPE * v;
        atomicMax(km + hd, fkey(v));
    }
}

// ============================================================
// Kernel 5: p = exp(logit - segmax); accumulate segsum and
//           UNNORMALIZED aggregate  Σ p * h[src]
// ============================================================
__global__ void gat_edge_acc(const long long* __restrict__ ei, int E, int N,
                             const float* __restrict__ asrc,
                             const float* __restrict__ adst,
                             const unsigned* __restrict__ kmax,
                             const float* __restrict__ h,
                             float* __restrict__ segsum,
                             float* __restrict__ agg) {
    const int e = (int)(blockIdx.x * blockDim.x + threadIdx.x);
    const int Et = E + N;
    if (e >= Et) return;
    __builtin_prefetch(ei + e + 4096, 0, 0);
    int s, d;
    if (e < E) { s = (int)ei[e]; d = (int)ei[(size_t)E + e]; }
    else       { s = e - E; d = s; }
    const float4* As = (const float4*)(asrc + (size_t)s * HEADS);
    const float4* Ad = (const float4*)(adst + (size_t)d * HEADS);
    float4 s0 = As[0], s1 = As[1], d0 = Ad[0], d1 = Ad[1];
    float l[HEADS] = { s0.x + d0.x, s0.y + d0.y, s0.z + d0.z, s0.w + d0.w,
                       s1.x + d1.x, s1.y + d1.y, s1.z + d1.z, s1.w + d1.w };
    const float4* hp = (const float4*)(h + (size_t)s * HC);
    float hv[HC];
    #pragma unroll
    for (int q = 0; q < 4; ++q) {
        float4 v = hp[q];
        hv[4*q+0] = v.x; hv[4*q+1] = v.y; hv[4*q+2] = v.z; hv[4*q+3] = v.w;
    }
    const unsigned* km = kmax + (size_t)d * HEADS;
    float* ss = segsum + (size_t)d * HEADS;
    float* ag = agg + (size_t)d * HC;
    #pragma unroll
    for (int hd = 0; hd < HEADS; ++hd) {
        float v = l[hd];
        v = (v > 0.0f) ? v : NEG_SLOPE * v;
        const float p = expf(v - funkey(km[hd]));
        atomicAdd(ss + hd, p);
        atomicAdd(ag + 2*hd + 0, p * hv[2*hd + 0]);
        atomicAdd(ag + 2*hd + 1, p * hv[2*hd + 1]);
    }
}

// ============================================================
// Kernel 6: normalize, add bias, apply 16x2 FC
// ============================================================
__global__ void gat_finalize(const float* __restrict__ agg,
                             const float* __restrict__ segsum,
                             const float* __restrict__ bias_gat,
                             const float* __restrict__ W_fc,
                             const float* __restrict__ b_fc,
                             float* __restrict__ out, int N) {
    const int n = (int)(blockIdx.x * blockDim.x + threadIdx.x);
    if (n >= N) return;
    float g[HC];
    #pragma unroll
    for (int hd = 0; hd < HEADS; ++hd) {
        const float inv = 1.0f / segsum[(size_t)n*HEADS + hd];
        g[2*hd + 0] = agg[(size_t)n*HC + 2*hd + 0] * inv + bias_gat[2*hd + 0];
        g[2*hd + 1] = agg[(size_t)n*HC + 2*hd + 1] * inv + bias_gat[2*hd + 1];
    }
    float o0 = b_fc[0], o1 = b_fc[1];
    #pragma unroll
    for (int i = 0; i < HC; ++i) {
        o0 = fmaf(g[i], W_fc[2*i + 0], o0);
        o1 = fmaf(g[i], W_fc[2*i + 1], o1);
    }
    out[(size_t)n*2 + 0] = o0;
    out[(size_t)n*2 + 1] = o1;
}

// ============================================================
// Host launcher
// ============================================================
extern "C" void kernel_launch(void* const* d_in, const int* in_sizes, int n_in,
                              void* d_out, int out_size, void* d_ws, size_t ws_size,
                              hipStream_t stream) {
    const float*     x        = (const float*)d_in[0];
    const long long* ei       = (const long long*)d_in[1];   // [2, E] int64
    // d_in[2] = edge_attr  (ignored, matches reference)
    const float*     W        = (const float*)d_in[3];
    const float*     att_src  = (const float*)d_in[4];
    const float*     att_dst  = (const float*)d_in[5];
    const float*     bias_gat = (const float*)d_in[6];
    const float*     W_fc     = (const float*)d_in[7];
    const float*     b_fc     = (const float*)d_in[8];
    float*           out      = (float*)d_out;

    const int N = in_sizes[0] / IN_F;
    const int E = in_sizes[1] / 2;
    const int Et = E + N;

    // workspace layout (floats): h[N*16] asrc[N*8] adst[N*8] kmax[N*8] segsum[N*8] agg[N*16]
    float*    h      = (float*)d_ws;
    float*    asrc   = h + (size_t)N * HC;
    float*    adst   = asrc + (size_t)N * HEADS;
    unsigned* kmax   = (unsigned*)(adst + (size_t)N * HEADS);
    float*    segsum = (float*)(kmax + (size_t)N * HEADS);
    float*    agg    = segsum + (size_t)N * HEADS;

    const int BLK = 256;

    // 1) h = x @ W  (WMMA): one wave per 16-row tile
    {
        const int waves  = (N + 15) / 16;
        const int blocks = (waves + (BLK / 32) - 1) / (BLK / 32);
        gat_gemm_wmma<<<blocks, BLK, 0, stream>>>(x, W, h, N);
    }
    // 2) per-node attention scalars
    gat_node_att<<<(N + BLK - 1) / BLK, BLK, 0, stream>>>(h, att_src, att_dst, asrc, adst, N);
    // 3) zero kmax + segsum + agg  (N*32 dwords = N*8 uint4)
    {
        const int nz4 = N * 8;
        gat_zero<<<(nz4 + BLK - 1) / BLK, BLK, 0, stream>>>((uint4*)kmax, nz4);
    }
    // 4) segment max
    gat_edge_max<<<(Et + BLK - 1) / BLK, BLK, 0, stream>>>(ei, E, N, asrc, adst, kmax);
    // 5) exp + accumulate
    gat_edge_acc<<<(Et + BLK - 1) / BLK, BLK, 0, stream>>>(ei, E, N, asrc, adst, kmax, h, segsum, agg);
    // 6) normalize + bias + FC
    gat_finalize<<<(N + BLK - 1) / BLK, BLK, 0, stream>>>(agg, segsum, bias_gat, W_fc, b_fc, out, N);
}